// NGCF_54984171323492
// MI455X (gfx1250) — compile-verified
//
#include <hip/hip_runtime.h>

// ---------------------------------------------------------------------------
// NGCF embedding propagation on MI455X (gfx1250, wave32).
//  - SpMM (memory-bound, ~1GB/layer) via float4 gathers + global f32 atomics.
//  - Dense 64-wide GEMMs via V_WMMA_F32_16X16X4_F32 (fp32 WMMA, exact precision;
//    GEMMs are a trivial FLOP fraction so no reason to down-convert).
//  - One wave32 per 16-row tile: EXEC is all-ones for WMMA, barriers are free.
// ---------------------------------------------------------------------------

#define N_USERS 30000
#define N_ITEMS 60000
#define N_NODES (N_USERS + N_ITEMS)   // 90000
#define NNZ     2000000
#define D       64
#define N_LAYERS 3
#define OUT_D   (D * (N_LAYERS + 1))  // 256
#define NTILES  (N_NODES / 16)        // 5625 (exact)
#define LDS_STRIDE 68                 // 64 + 4 pad -> conflict-free, 16B-aligned rows

typedef __attribute__((ext_vector_type(2))) float v2f;
typedef __attribute__((ext_vector_type(8))) float v8f;

// ---------------------------------------------------------------------------
// ego = concat(user_emb, item_emb); out[:, 0:64] = ego (raw, un-normalized).
// ---------------------------------------------------------------------------
__global__ void ngcf_init_kernel(const float* __restrict__ ue,
                                 const float* __restrict__ ie,
                                 float* __restrict__ ego,
                                 float* __restrict__ out) {
  int idx = blockIdx.x * blockDim.x + threadIdx.x;   // float4 index
  if (idx >= N_NODES * 16) return;
  int node = idx >> 4;
  int c4   = idx & 15;
  float4 v;
  if (node < N_USERS)
    v = *(const float4*)(ue + (long long)node * D + c4 * 4);
  else
    v = *(const float4*)(ie + (long long)(node - N_USERS) * D + c4 * 4);
  *(float4*)(ego + (long long)idx * 4) = v;
  *(float4*)(out + (long long)node * OUT_D + c4 * 4) = v;
}

// ---------------------------------------------------------------------------
// Zero fill (float4).
// ---------------------------------------------------------------------------
__global__ void ngcf_zero_kernel(float4* __restrict__ p, int n4) {
  int i = blockIdx.x * blockDim.x + threadIdx.x;
  if (i < n4) p[i] = make_float4(0.f, 0.f, 0.f, 0.f);
}

// ---------------------------------------------------------------------------
// side[row] += val * ego[col]  (COO scatter).  One thread per (edge, float4).
// 16 consecutive threads share an edge -> 256B coalesced gather from ego
// (23MB table: L2-resident on the 192MB L2). Scatter via f32 atomics.
// ---------------------------------------------------------------------------
__global__ void ngcf_spmm_kernel(const int* __restrict__ adj_row,
                                 const int* __restrict__ adj_col,
                                 const float* __restrict__ adj_val,
                                 const float* __restrict__ ego,
                                 float* __restrict__ side) {
  int t = blockIdx.x * blockDim.x + threadIdx.x;
  if (t >= NNZ * 16) return;
  int e   = t >> 4;
  int seg = t & 15;
  int r = adj_row[e];
  int c = adj_col[e];
  float v = adj_val[e];
  float4 s = *(const float4*)(ego + (long long)c * D + seg * 4);
  float* dst = side + (long long)r * D + seg * 4;
  atomicAdd(dst + 0, s.x * v);
  atomicAdd(dst + 1, s.y * v);
  atomicAdd(dst + 2, s.z * v);
  atomicAdd(dst + 3, s.w * v);
}

// ---------------------------------------------------------------------------
// Fused layer: for a 16-row tile,
//   sR = lrelu(side @ Wgc + bgc) + lrelu((ego*side) @ Wbi + bbi)
//   ego <- sR ; out[:, (layer+1)*64 : ...] <- l2_normalize_rows(sR)
// One wave32 per block/tile; WMMA fragments per ISA 7.12.2:
//   A: lane m=l&15 holds row m; VGPR v holds K = 4s + 2*(l>>4) + v
//   B: lane n=l&15 holds col n; VGPR v holds K = 4s + 2*(l>>4) + v
//   C/D: VGPR r holds row r + 8*(l>>4), col l&15
// ---------------------------------------------------------------------------
__global__ __launch_bounds__(32) void ngcf_layer_kernel(
    const float* __restrict__ side,
    float* ego,                         // read own tile, then overwritten
    const float* __restrict__ Wgc, const float* __restrict__ bgc,
    const float* __restrict__ Wbi, const float* __restrict__ bbi,
    float* __restrict__ out, int layer) {
  __shared__ __align__(16) float sA[16 * LDS_STRIDE];  // side tile
  __shared__ __align__(16) float sP[16 * LDS_STRIDE];  // ego*side tile
  __shared__ __align__(16) float sR[16 * LDS_STRIDE];  // result tile
  __shared__ float sScale[16];

  const int lane = threadIdx.x;
  const int tile = blockIdx.x;
  const int m  = lane & 15;
  const int hl = lane >> 4;
  const long long row0 = (long long)tile * 16;

  // W matrices are 16KB each and shared by all 5625 blocks: keep them hot.
  __builtin_prefetch(Wgc, 0, 3);
  __builtin_prefetch(Wbi, 0, 3);

  // Stage tiles into LDS (8 x float4 per lane, coalesced).
  #pragma unroll
  for (int i = 0; i < 8; ++i) {
    int idx = i * 32 + lane;
    int r = idx >> 4, c4 = idx & 15;
    float4 sv = *(const float4*)(side + (row0 + r) * D + c4 * 4);
    float4 ev = *(const float4*)(ego  + (row0 + r) * D + c4 * 4);
    *(float4*)&sA[r * LDS_STRIDE + c4 * 4] = sv;
    float4 pv = make_float4(sv.x * ev.x, sv.y * ev.y, sv.z * ev.z, sv.w * ev.w);
    *(float4*)&sP[r * LDS_STRIDE + c4 * 4] = pv;
  }
  __syncthreads();

  // ---- GEMM1: sR = lrelu(sA @ Wgc + bgc) ----
  #pragma unroll
  for (int t = 0; t < 4; ++t) {
    float bv = bgc[t * 16 + m];
    v8f acc;
    #pragma unroll
    for (int r = 0; r < 8; ++r) acc[r] = bv;
    #pragma unroll
    for (int s = 0; s < 16; ++s) {
      int k = 4 * s + 2 * hl;
      v2f a = *(const v2f*)&sA[m * LDS_STRIDE + k];   // 8B-aligned ds_load_b64
      v2f b;
      b[0] = Wgc[(k + 0) * D + t * 16 + m];
      b[1] = Wgc[(k + 1) * D + t * 16 + m];
      acc = __builtin_amdgcn_wmma_f32_16x16x4_f32(false, a, false, b,
                                                  (short)0, acc, false, false);
    }
    #pragma unroll
    for (int r = 0; r < 8; ++r) {
      float x = acc[r];
      x = x > 0.0f ? x : 0.2f * x;
      sR[(r + 8 * hl) * LDS_STRIDE + t * 16 + m] = x;
    }
  }
  __syncthreads();

  // ---- GEMM2: sR += lrelu(sP @ Wbi + bbi) ----
  #pragma unroll
  for (int t = 0; t < 4; ++t) {
    float bv = bbi[t * 16 + m];
    v8f acc;
    #pragma unroll
    for (int r = 0; r < 8; ++r) acc[r] = bv;
    #pragma unroll
    for (int s = 0; s < 16; ++s) {
      int k = 4 * s + 2 * hl;
      v2f a = *(const v2f*)&sP[m * LDS_STRIDE + k];
      v2f b;
      b[0] = Wbi[(k + 0) * D + t * 16 + m];
      b[1] = Wbi[(k + 1) * D + t * 16 + m];
      acc = __builtin_amdgcn_wmma_f32_16x16x4_f32(false, a, false, b,
                                                  (short)0, acc, false, false);
    }
    #pragma unroll
    for (int r = 0; r < 8; ++r) {
      float x = acc[r];
      x = x > 0.0f ? x : 0.2f * x;
      sR[(r + 8 * hl) * LDS_STRIDE + t * 16 + m] += x;
    }
  }
  __syncthreads();

  // ---- Row L2 norms: lane (m, hl) sums half a row, combine across half-waves.
  float ss = 0.0f;
  #pragma unroll
  for (int c = 0; c < 32; ++c) {
    float x = sR[m * LDS_STRIDE + hl * 32 + c];
    ss += x * x;
  }
  ss += __shfl_xor(ss, 16, 32);
  if (hl == 0) sScale[m] = 1.0f / fmaxf(sqrtf(ss), 1e-12f);
  __syncthreads();

  // ---- Write ego (raw) and normalized output slab (float4, coalesced).
  const int col0 = (layer + 1) * D;
  #pragma unroll
  for (int i = 0; i < 8; ++i) {
    int idx = i * 32 + lane;
    int r = idx >> 4, c4 = idx & 15;
    float4 v = *(const float4*)&sR[r * LDS_STRIDE + c4 * 4];
    long long node = row0 + r;
    *(float4*)(ego + node * D + c4 * 4) = v;
    float sc = sScale[r];
    float4 vn = make_float4(v.x * sc, v.y * sc, v.z * sc, v.w * sc);
    *(float4*)(out + node * OUT_D + col0 + c4 * 4) = vn;
  }
}

// ---------------------------------------------------------------------------
extern "C" void kernel_launch(void* const* d_in, const int* in_sizes, int n_in,
                              void* d_out, int out_size, void* d_ws, size_t ws_size,
                              hipStream_t stream) {
  (void)in_sizes; (void)n_in; (void)out_size; (void)ws_size;

  const float* user_emb = (const float*)d_in[0];   // [30000,64]
  const float* item_emb = (const float*)d_in[1];   // [60000,64]
  const int*   adj_row  = (const int*)d_in[2];     // [2M]
  const int*   adj_col  = (const int*)d_in[3];     // [2M]
  const float* adj_val  = (const float*)d_in[4];   // [2M]
  const float* W_gc     = (const float*)d_in[5];   // [3,64,64]
  const float* b_gc     = (const float*)d_in[6];   // [3,1,64]
  const float* W_bi     = (const float*)d_in[7];   // [3,64,64]
  const float* b_bi     = (const float*)d_in[8];   // [3,1,64]
  float* out = (float*)d_out;                      // [90000,256] flat

  // Workspace: ego [90000,64] + side [90000,64] = 46.1 MB.
  float* ego  = (float*)d_ws;
  float* side = ego + (size_t)N_NODES * D;

  // ego = concat(user,item); out slab 0 = ego.
  {
    int n4 = N_NODES * 16;
    ngcf_init_kernel<<<(n4 + 255) / 256, 256, 0, stream>>>(user_emb, item_emb,
                                                           ego, out);
  }

  for (int k = 0; k < N_LAYERS; ++k) {
    int n4 = N_NODES * 16;
    ngcf_zero_kernel<<<(n4 + 255) / 256, 256, 0, stream>>>((float4*)side, n4);

    int nt = NNZ * 16;
    ngcf_spmm_kernel<<<(nt + 255) / 256, 256, 0, stream>>>(adj_row, adj_col,
                                                           adj_val, ego, side);

    ngcf_layer_kernel<<<NTILES, 32, 0, stream>>>(
        side, ego,
        W_gc + (size_t)k * D * D, b_gc + (size_t)k * D,
        W_bi + (size_t)k * D * D, b_bi + (size_t)k * D,
        out, k);
  }
}